// MoEGating_55405078119404
// MI455X (gfx1250) — compile-verified
//
#include <hip/hip_runtime.h>

#define D 2048
#define E 64
#define KC 32
#define KPAD 36                 // tile row stride (floats): 16B-aligned, conflict-free
#define TPAD 20                 // transpose token stride (floats): 16B-aligned
#define TOK_PER_WG 64           // 4 waves * 16 tokens
#define NCHUNK (D / KC)

typedef float v2f __attribute__((ext_vector_type(2)));
typedef float v8f __attribute__((ext_vector_type(8)));
typedef unsigned long long u64;

// Async copy 16 bytes global -> LDS, tracked by ASYNCcnt (CDNA5).
// GV mode: vdst = LDS byte offset (VGPR), vaddr = 64-bit global address.
__device__ __forceinline__ void async_ld_b128(unsigned lds_off, const void* gptr) {
    asm volatile("global_load_async_to_lds_b128 %0, %1, off"
                 :: "v"(lds_off), "v"(gptr)
                 : "memory");
}

__device__ __forceinline__ void wait_async0() {
    asm volatile("s_wait_asynccnt 0x0" ::: "memory");
}

__global__ __launch_bounds__(128) void moe_gate_topk_kernel(
    const float* __restrict__ x, const float* __restrict__ W,
    int* __restrict__ idx_out, float* __restrict__ val_out)
{
    __shared__ float Xl[2][TOK_PER_WG][KPAD];  // 64 tokens x KC (double buffered)
    __shared__ float Wl[2][E][KPAD];           // 64 experts x KC
    __shared__ float Tl[4][E][TPAD];           // per-wave logit transpose scratch

    const int tid  = threadIdx.x;
    const int lane = tid & 31;
    const int wv   = tid >> 5;                 // wave id 0..3
    const int tok0 = blockIdx.x * TOK_PER_WG;

    // ---- cooperative async loader: 128 threads, b128 granularity ----
    const int lrow = tid >> 3;                 // 0..15
    const int lcol = (tid & 7) * 4;            // 0,4,...,28  (covers KC=32)

    const float* xg = x + (size_t)(tok0 + lrow) * D + lcol;
    const float* wg = W + (size_t)lrow * D + lcol;

    auto issue_async = [&](int k0, int buf) {
#pragma unroll
        for (int i = 0; i < 4; ++i) {          // rows lrow, +16, +32, +48
            const unsigned lx = (unsigned)(size_t)&Xl[buf][lrow + 16 * i][lcol];
            const unsigned lw = (unsigned)(size_t)&Wl[buf][lrow + 16 * i][lcol];
            async_ld_b128(lx, xg + (size_t)(16 * i) * D + k0);
            async_ld_b128(lw, wg + (size_t)(16 * i) * D + k0);
        }
    };

    // ---- WMMA fragment addressing (ISA 16x16x4 f32 layouts) ----
    const int hrow = lane & 15;                // A: token row / B: expert col
    const int half = lane >> 4;
    const int khi  = half * 2;                 // K sub-pair select

    v8f acc[4] = {};                           // 4 expert tiles of 16x16 f32

    issue_async(0, 0);
    wait_async0();
    __syncthreads();

    for (int ch = 0; ch < NCHUNK; ++ch) {
        const int buf = ch & 1;
        if (ch + 1 < NCHUNK) issue_async((ch + 1) * KC, (ch + 1) & 1);

        const float* xa = &Xl[buf][wv * 16 + hrow][khi];
#pragma unroll
        for (int kk = 0; kk < KC; kk += 4) {
            v2f a  = *(const v2f*)(xa + kk);
            v2f b0 = *(const v2f*)&Wl[buf][ 0 + hrow][kk + khi];
            v2f b1 = *(const v2f*)&Wl[buf][16 + hrow][kk + khi];
            v2f b2 = *(const v2f*)&Wl[buf][32 + hrow][kk + khi];
            v2f b3 = *(const v2f*)&Wl[buf][48 + hrow][kk + khi];
            acc[0] = __builtin_amdgcn_wmma_f32_16x16x4_f32(false, a, false, b0,
                         (short)0, acc[0], false, false);
            acc[1] = __builtin_amdgcn_wmma_f32_16x16x4_f32(false, a, false, b1,
                         (short)0, acc[1], false, false);
            acc[2] = __builtin_amdgcn_wmma_f32_16x16x4_f32(false, a, false, b2,
                         (short)0, acc[2], false, false);
            acc[3] = __builtin_amdgcn_wmma_f32_16x16x4_f32(false, a, false, b3,
                         (short)0, acc[3], false, false);
        }

        wait_async0();      // our async writes into buf^1 have landed in LDS
        __syncthreads();    // everyone done computing buf + sees new buf^1
    }

    // ---- epilogue: transpose logits via LDS, packed-key top-2, softmax ----
    // C layout: lane holds expert e = hrow + 16*et, tokens m = 8*half + r.
    // Store [expert][token] (token stride TPAD): two aligned b128 per tile.
#pragma unroll
    for (int et = 0; et < 4; ++et) {
        const int e = hrow + 16 * et;
        float* dst = &Tl[wv][e][8 * half];
        *(float4*)(dst + 0) = make_float4(acc[et][0], acc[et][1], acc[et][2], acc[et][3]);
        *(float4*)(dst + 4) = make_float4(acc[et][4], acc[et][5], acc[et][6], acc[et][7]);
    }
    // Same-wave producer/consumer; LDS ops are in-order per wave -> no barrier.

    // Two lanes per token: lane = half*16 + t scans experts e = 2*i + half.
    // key = (monotone(value) << 6) | (63 - e): u64 compare == value order with
    // exact top_k tie-break (lower index wins); keys are unique.
    const int t = hrow;
    u64 s1 = 0, s2 = 0;
#pragma unroll
    for (int i = 0; i < 32; ++i) {
        const int e = 2 * i + half;
        const unsigned b = __float_as_uint(Tl[wv][e][t]);
        const unsigned u = b ^ (unsigned)(((int)b >> 31) | 0x80000000);
        const u64 k = ((u64)u << 6) | (unsigned)(63 - e);
        if (k > s1)      { s2 = s1; s1 = k; }
        else if (k > s2) { s2 = k; }
    }
    {   // merge the two half-lane partials (xor 16)
        const u64 o1 = __shfl_xor(s1, 16, 32);
        const u64 o2 = __shfl_xor(s2, 16, 32);
        if (o1 > s1)      { s2 = s1; s1 = o1; }
        else if (o1 > s2) { s2 = o1; }
        if (o2 > s1)      { s2 = s1; s1 = o2; }
        else if (o2 > s2) { s2 = o2; }
    }

    if (half == 0) {                           // lanes 0..15 -> 16 tokens, coalesced
        const int e1 = 63 - (int)(s1 & 63);
        const int e2 = 63 - (int)(s2 & 63);
        const unsigned u1 = (unsigned)(s1 >> 6);
        const unsigned u2 = (unsigned)(s2 >> 6);
        const float v1 = __uint_as_float((u1 & 0x80000000u) ? (u1 ^ 0x80000000u) : ~u1);
        const float v2 = __uint_as_float((u2 & 0x80000000u) ? (u2 ^ 0x80000000u) : ~u2);

        const int tk = tok0 + wv * 16 + t;
        idx_out[2 * tk + 0] = e1;
        idx_out[2 * tk + 1] = e2;
        const float eo  = __expf(v2 - v1);     // <= 1, stable
        const float inv = 1.0f / (1.0f + eo);
        val_out[2 * tk + 0] = inv;
        val_out[2 * tk + 1] = eo * inv;
    }
}

extern "C" void kernel_launch(void* const* d_in, const int* in_sizes, int n_in,
                              void* d_out, int out_size, void* d_ws, size_t ws_size,
                              hipStream_t stream) {
    const float* x = (const float*)d_in[0];
    const float* W = (const float*)d_in[1];
    const int N = in_sizes[0] / D;                 // 16384 tokens
    int*   idx_out = (int*)d_out;                  // first output: int32 [N,2]
    float* val_out = (float*)d_out + (size_t)N * 2;// second output: f32 [N,2]

    dim3 grid(N / TOK_PER_WG);
    dim3 block(128);
    hipLaunchKernelGGL(moe_gate_topk_kernel, grid, block, 0, stream,
                       x, W, idx_out, val_out);
}